// EdgeGuidedAttention_18296560681701
// MI455X (gfx1250) — compile-verified
//
#include <hip/hip_runtime.h>
#include <hip/hip_bf16.h>

// B=4, H=8, N=256, D=64
#define NB 4
#define NH 8
#define NN 256
#define ND 64
#define NEG_INF_V (-1000000000.0f)

typedef __attribute__((ext_vector_type(2))) float v2f;
typedef __attribute__((ext_vector_type(8))) float v8f;
typedef unsigned int u32x4 __attribute__((ext_vector_type(4)));
typedef int i32x4 __attribute__((ext_vector_type(4)));
typedef int i32x8 __attribute__((ext_vector_type(8)));

#if defined(__has_builtin)
#if __has_builtin(__builtin_amdgcn_tensor_load_to_lds) && __has_builtin(__builtin_amdgcn_s_wait_tensorcnt)
#define USE_TDM 1
#endif
#endif
#ifndef USE_TDM
#define USE_TDM 0
#endif

#if USE_TDM
// 2D TDM load: global tile (tile_d0 cols x tile_d1 rows, row stride stride_d0,
// 4-byte elements) -> LDS at lds_dst, padding pad (code) dwords after every
// 2^(pad_int_code+1) dwords stored (ISA ch8: D# group0/group1 bitfields).
__device__ __forceinline__ void tdm_load_2d(void* lds_dst, const void* gsrc,
                                            unsigned tile_d0, unsigned tile_d1,
                                            unsigned stride_d0,
                                            unsigned pad_int_code,
                                            unsigned pad_amt_code) {
  unsigned long long ga = (unsigned long long)(uintptr_t)gsrc;
  unsigned lds = (unsigned)(uintptr_t)lds_dst;   // low 32 bits of generic = LDS offset
  u32x4 g0;
  g0[0] = 1u;                                    // count=1 (valid descriptor)
  g0[1] = lds;                                   // lds_addr
  g0[2] = (unsigned)ga;                          // global_addr[31:0]
  g0[3] = (unsigned)(ga >> 32) | (2u << 30);     // global_addr[56:32] | type=2
  i32x8 g1;
  g1[0] = (int)((2u << 16)                       // data_size = 4B
              | (1u << 20)                       // pad_enable
              | (pad_int_code << 22)             // pad interval code
              | (pad_amt_code << 25));           // pad amount code
  g1[1] = (int)((tile_d0 & 0xFFFFu) << 16);      // tensor_dim0[15:0] @ bits 63:48
  g1[2] = (int)(((tile_d0 >> 16) & 0xFFFFu)      // tensor_dim0[31:16]
              | ((tile_d1 & 0xFFFFu) << 16));    // tensor_dim1[15:0]
  g1[3] = (int)(((tile_d1 >> 16) & 0xFFFFu)      // tensor_dim1[31:16]
              | ((tile_d0 & 0xFFFFu) << 16));    // tile_dim0
  g1[4] = (int)(tile_d1 & 0xFFFFu);              // tile_dim1 (tile_dim2=0)
  g1[5] = (int)stride_d0;                        // tensor_dim0_stride[31:0]
  g1[6] = 0;                                     // stride hi / dim1_stride lo
  g1[7] = 0;                                     // dim1_stride hi (unused, 2D)
  i32x4 z4 = {0, 0, 0, 0};
#if __clang_major__ >= 23
  i32x8 z8 = {0, 0, 0, 0, 0, 0, 0, 0};
  __builtin_amdgcn_tensor_load_to_lds(g0, g1, z4, z4, z8, 0);
#else
  __builtin_amdgcn_tensor_load_to_lds(g0, g1, z4, z4, 0);
#endif
}
#endif

// ---------------------------------------------------------------------------
// Kernel A: projections via V_WMMA_F32_16X16X4_F32.
//   src==0: a[bh][n][e] = sum_d Q[bh][n][d]*W_h[e][d]      + b_h[e]
//   src==1: c[bh][m][e] = sum_d K[bh][m][d]*W_h[e][64+d]
// ---------------------------------------------------------------------------
__global__ __launch_bounds__(128)
void proj_wmma_kernel(const float* __restrict__ Q, const float* __restrict__ K,
                      const float* __restrict__ Wh, const float* __restrict__ bh_,
                      float* __restrict__ Aout, float* __restrict__ Cout) {
  __shared__ __align__(16) float qS[256 * 68];   // 256 rows of 64, pad -> 68
  __shared__ __align__(16) float wS[64 * 132];   // 64 rows of 128, pad -> 132

  const int blk  = blockIdx.x;        // 0..63
  const int src  = blk & 1;
  const int bhid = blk >> 1;          // 0..31
  const int t    = threadIdx.x;       // 0..127

  const float* S = (src == 0 ? Q : K) + bhid * (NN * ND);

#if USE_TDM
  if (t < 32) {
    tdm_load_2d(qS, S, 64, 256, 64, /*64 dwords*/5, /*+4*/3);
    tdm_load_2d(wS, Wh, 128, 64, 128, /*128 dwords*/6, /*+4*/3);
    __builtin_amdgcn_s_wait_tensorcnt(0);
  }
#else
  #pragma unroll
  for (int i = 0; i < 32; ++i) {
    int f4 = t + i * 128;
    int m  = f4 >> 4;
    int e  = (f4 & 15) << 2;
    *(float4*)&qS[m * 68 + e] = *(const float4*)&S[f4 * 4];
  }
  #pragma unroll
  for (int i = 0; i < 16; ++i) {
    int f4 = t + i * 128;
    int e  = f4 >> 5;
    int d  = (f4 & 31) << 2;
    *(float4*)&wS[e * 132 + d] = *(const float4*)&Wh[f4 * 4];
  }
#endif
  __syncthreads();

  const int wave = t >> 5;
  const int lane = t & 31;
  const int l15  = lane & 15;
  const int half = lane >> 4;

  float* Dst = (src == 0 ? Aout : Cout) + bhid * (NN * ND);

  for (int tt = 0; tt < 16; ++tt) {
    int tile = wave * 16 + tt;            // 0..63
    int n0 = (tile >> 2) * 16;
    int e0 = (tile & 3) * 16;
    v8f acc = {};
    #pragma unroll
    for (int kk = 0; kk < ND; kk += 4) {
      v2f a = *(const v2f*)&qS[(n0 + l15) * 68 + kk + 2 * half];
      v2f b;
      b.x = wS[(e0 + l15) * 132 + src * 64 + kk + 2 * half + 0];
      b.y = wS[(e0 + l15) * 132 + src * 64 + kk + 2 * half + 1];
      acc = __builtin_amdgcn_wmma_f32_16x16x4_f32(false, a, false, b, (short)0, acc, false, false);
    }
    float bias = (src == 0) ? bh_[e0 + l15] : 0.0f;
    #pragma unroll
    for (int r = 0; r < 8; ++r)
      Dst[(n0 + r + 8 * half) * ND + e0 + l15] = acc[r] + bias;
  }
}

// ---------------------------------------------------------------------------
// Kernel B: scores + mask + softmax.
// leaky_relu(x) = 0.6x + 0.4|x|  -> per (n,m,e): 1 add + 2 FMA (|x| = src mod).
// ---------------------------------------------------------------------------
__global__ __launch_bounds__(256)
void scores_softmax_kernel(const float* __restrict__ Aw, const float* __restrict__ Cw,
                           const float* __restrict__ Wo, const float* __restrict__ bo,
                           const int* __restrict__ mask, float* __restrict__ Pout) {
  __shared__ __align__(16) float cS[256 * 68];
  __shared__ __align__(16) float aS[16 * 68];
  __shared__ __align__(16) float scS[16 * 260];
  __shared__ __align__(16) float w6S[64];
  __shared__ __align__(16) float w4S[64];

  const int blk  = blockIdx.x;          // 0..511
  const int nt   = blk & 15;
  const int bhid = blk >> 4;            // 0..31
  const int b    = bhid >> 3;           // batch (H=8)
  const int t    = threadIdx.x;         // 0..255

  const float* Csrc = Cw + bhid * (NN * ND);
  const float* Asrc = Aw + bhid * (NN * ND) + nt * 16 * ND;

#if USE_TDM
  if (t < 32) {
    tdm_load_2d(cS, Csrc, 64, 256, 64, 5, 3);
    tdm_load_2d(aS, Asrc, 64, 16, 64, 5, 3);
    __builtin_amdgcn_s_wait_tensorcnt(0);
  }
#else
  #pragma unroll
  for (int i = 0; i < 16; ++i) {
    int f4 = t + i * 256;
    int m  = f4 >> 4;
    int e  = (f4 & 15) << 2;
    *(float4*)&cS[m * 68 + e] = *(const float4*)&Csrc[f4 * 4];
  }
  {
    int n = t >> 4;
    int e = (t & 15) << 2;
    *(float4*)&aS[n * 68 + e] = *(const float4*)&Asrc[t * 4];
  }
#endif
  if (t < 64) {
    float w = Wo[t];
    w6S[t] = 0.6f * w;
    w4S[t] = 0.4f * w;
  }
  const float bov = bo[0];
  __syncthreads();

  const int m  = t;
  const int n0 = nt * 16;
  float s[16];
  #pragma unroll
  for (int n = 0; n < 16; ++n) s[n] = 0.0f;

  #pragma unroll 2
  for (int c = 0; c < 16; ++c) {
    const int e0 = c * 4;
    float4 c4 = *(const float4*)&cS[m * 68 + e0];
    float4 w6 = *(const float4*)&w6S[e0];
    float4 w4 = *(const float4*)&w4S[e0];
    #pragma unroll
    for (int n = 0; n < 16; ++n) {
      float4 a4 = *(const float4*)&aS[n * 68 + e0];   // wave-broadcast read
      float h0 = a4.x + c4.x;
      float h1 = a4.y + c4.y;
      float h2 = a4.z + c4.z;
      float h3 = a4.w + c4.w;
      s[n] += w6.x * h0 + w4.x * __builtin_fabsf(h0);
      s[n] += w6.y * h1 + w4.y * __builtin_fabsf(h1);
      s[n] += w6.z * h2 + w4.z * __builtin_fabsf(h2);
      s[n] += w6.w * h3 + w4.w * __builtin_fabsf(h3);
    }
  }

  const int* mrow = mask + b * (NN * NN) + n0 * NN + m;
  #pragma unroll
  for (int n = 0; n < 16; ++n) {
    int mk = mrow[n * NN];
    scS[n * 260 + m] = (mk == 0) ? NEG_INF_V : (s[n] + bov);
  }
  __syncthreads();

  // Softmax: 8 waves, wave handles rows 2w, 2w+1; lane covers 8 columns.
  const int wave = t >> 5;
  const int lane = t & 31;
  float* prow_base = Pout + bhid * (NN * NN) + n0 * NN;
  #pragma unroll
  for (int rr = 0; rr < 2; ++rr) {
    int r = wave * 2 + rr;
    float v[8];
    float mx = -3.4e38f;
    #pragma unroll
    for (int j = 0; j < 8; ++j) {
      v[j] = scS[r * 260 + lane + 32 * j];
      mx = fmaxf(mx, v[j]);
    }
    #pragma unroll
    for (int off = 16; off > 0; off >>= 1) mx = fmaxf(mx, __shfl_xor(mx, off, 32));
    float sum = 0.0f;
    #pragma unroll
    for (int j = 0; j < 8; ++j) { v[j] = __expf(v[j] - mx); sum += v[j]; }
    #pragma unroll
    for (int off = 16; off > 0; off >>= 1) sum += __shfl_xor(sum, off, 32);
    float inv = 1.0f / sum;
    #pragma unroll
    for (int j = 0; j < 8; ++j)
      prow_base[r * NN + lane + 32 * j] = v[j] * inv;
  }
}

// ---------------------------------------------------------------------------
// Kernel C: out = P @ V via V_WMMA_F32_16X16X4_F32.
// Block = 256 thr (8 waves), fixed bh; TDM-stages V (256x64) and a 32-row
// P slab; each wave computes one 16x16 output tile, K=256 -> 64 wmma steps.
// ---------------------------------------------------------------------------
__global__ __launch_bounds__(256)
void pv_wmma_kernel(const float* __restrict__ P, const float* __restrict__ V,
                    float* __restrict__ O) {
  __shared__ __align__(16) float vS[256 * 68];    // V rows m, pad -> 68
  __shared__ __align__(16) float pS[32 * 260];    // 32 P rows, pad -> 260

  const int blk  = blockIdx.x;          // 0..255
  const int bhid = blk >> 3;            // 0..31
  const int slab = blk & 7;             // 8 slabs of 32 rows
  const int t    = threadIdx.x;

  const float* Vsrc = V + bhid * (NN * ND);
  const float* Psrc = P + bhid * (NN * NN) + slab * 32 * NN;

#if USE_TDM
  if (t < 32) {
    tdm_load_2d(vS, Vsrc, 64, 256, 64, /*64 dw*/5, /*+4*/3);
    tdm_load_2d(pS, Psrc, 256, 32, 256, /*256 dw*/7, /*+4*/3);
    __builtin_amdgcn_s_wait_tensorcnt(0);
  }
#else
  #pragma unroll
  for (int i = 0; i < 16; ++i) {
    int f4 = t + i * 256;
    int m  = f4 >> 4;
    int e  = (f4 & 15) << 2;
    *(float4*)&vS[m * 68 + e] = *(const float4*)&Vsrc[f4 * 4];
  }
  #pragma unroll
  for (int i = 0; i < 8; ++i) {
    int f4 = t + i * 256;                 // 0..2047 float4s (32x256 floats)
    int r  = f4 >> 6;
    int c  = (f4 & 63) << 2;
    *(float4*)&pS[r * 260 + c] = *(const float4*)&Psrc[f4 * 4];
  }
#endif
  __syncthreads();

  const int wave = t >> 5;
  const int lane = t & 31;
  const int l15  = lane & 15;
  const int half = lane >> 4;
  const int dt   = wave & 3;            // D col tile
  const int rh   = wave >> 2;           // which 16-row half of the slab

  v8f acc = {};
  for (int kk = 0; kk < NN; kk += 4) {
    v2f a = *(const v2f*)&pS[(rh * 16 + l15) * 260 + kk + 2 * half];
    v2f bf;
    bf.x = vS[(kk + 2 * half + 0) * 68 + dt * 16 + l15];
    bf.y = vS[(kk + 2 * half + 1) * 68 + dt * 16 + l15];
    acc = __builtin_amdgcn_wmma_f32_16x16x4_f32(false, a, false, bf, (short)0, acc, false, false);
  }
  float* o = O + bhid * (NN * ND) + (slab * 32 + rh * 16) * ND + dt * 16;
  #pragma unroll
  for (int r = 0; r < 8; ++r)
    o[(r + 8 * half) * ND + l15] = acc[r];
}

// ---------------------------------------------------------------------------
extern "C" void kernel_launch(void* const* d_in, const int* in_sizes, int n_in,
                              void* d_out, int out_size, void* d_ws, size_t ws_size,
                              hipStream_t stream) {
  const float* query = (const float*)d_in[0];
  const float* key_  = (const float*)d_in[1];
  const float* value = (const float*)d_in[2];
  const int*   mask  = (const int*)d_in[3];
  const float* W_h   = (const float*)d_in[4];
  const float* b_h   = (const float*)d_in[5];
  const float* W_o   = (const float*)d_in[6];
  const float* b_o   = (const float*)d_in[7];

  float* out  = (float*)d_out;                       // [4,8,256,64]  = 524288
  float* patt = out + (NB * NH * NN * ND);           // [4,8,256,256] = 2097152

  float* wsA = (float*)d_ws;                         // a = q_proj + b_h
  float* wsC = wsA + (NB * NH * NN * ND);            // c = k_proj

  proj_wmma_kernel<<<NB * NH * 2, 128, 0, stream>>>(query, key_, W_h, b_h, wsA, wsC);
  scores_softmax_kernel<<<NB * NH * 16, 256, 0, stream>>>(wsA, wsC, W_o, b_o, mask, patt);
  pv_wmma_kernel<<<(NB * NH * 16 * 4) / 8, 256, 0, stream>>>(patt, value, out);
}